// Margin_Cosine_Similarity_Loss_1958505087537
// MI455X (gfx1250) — compile-verified
//
#include <hip/hip_runtime.h>

typedef __attribute__((ext_vector_type(2))) float v2f;
typedef __attribute__((ext_vector_type(8))) float v8f;

#define NCLS 10
#define NEGINF (-3.402823466e38f)

// D[class][sample] = emb[16x12pad] x x1^T[12pad x 16] via V_WMMA_F32_16X16X4_F32 (3 k-steps).
// Lane pair (n, n+16) owns sample n of the tile: lane n has classes 0-7 in d[0..7],
// lane n+16 has classes 8-9 in d[0..1] (rest masked to -inf). Top-2 is lane-local + 1 merge.
__global__ __launch_bounds__(256) void mcsl_main(
    const float* __restrict__ x1, const int* __restrict__ tgt,
    const float* __restrict__ emb, float* __restrict__ partials,
    int batch, int ntiles)
{
  const int lane = threadIdx.x & 31;
  const int wid  = threadIdx.x >> 5;
  const int half = lane >> 4;   // K-pair / D-row-half selector
  const int nidx = lane & 15;   // class index for A(emb); sample-in-tile for B(x1)
  const int koff = half * 2;

  // ---- A fragments from emb (constant per wave), ISA 16x4 f32 A layout:
  //      lane l: row = l%16, VGPR0/1 = K = (l/16)*2 + {0,1} within each 4-wide k-step.
  v2f afrag[3];
  {
    const int c  = nidx;
    const int cc = (c < NCLS) ? c : 0;
#pragma unroll
    for (int s = 0; s < 3; ++s) {
      int kb = 4 * s + koff;
      int kc = (kb <= 8) ? kb : 8;           // clamp address, mask value
      v2f ev = *(const v2f*)(emb + cc * NCLS + kc);
      float m = ((c < NCLS) && (kb <= 8)) ? 1.f : 0.f;
      afrag[s].x = ev.x * m;
      afrag[s].y = ev.y * m;
    }
  }
  // per-class 1/||emb[c]|| held at lane c (and c+16)
  float inv_embn;
  {
    const int cc = (nidx < NCLS) ? nidx : 0;
    float s2 = 0.f;
#pragma unroll
    for (int k = 0; k < NCLS; ++k) { float e = emb[cc * NCLS + k]; s2 += e * e; }
    inv_embn = 1.0f / sqrtf(s2);
  }

  const int nwaves = gridDim.x * (blockDim.x >> 5);
  const int wave   = blockIdx.x * (blockDim.x >> 5) + wid;

  float acc = 0.f;

  for (int tile = wave; tile < ntiles; tile += nwaves) {
    const int   base  = tile * 16;
    const int   srow  = base + nidx;
    const int   rowc  = (srow < batch) ? srow : (batch - 1);
    const float rmask = (srow < batch) ? 1.f : 0.f;

    // ---- B fragments (x1 used transposed) + per-sample sum of squares.
    //      B 4x16 layout mirrors A: lane l supplies K = (l/16)*2 + {0,1}, N = l%16.
    //      Each lane -> one aligned b64 load per k-step.
    v2f bfrag[3];
    float sq = 0.f;
#pragma unroll
    for (int s = 0; s < 3; ++s) {
      int kb = 4 * s + koff;
      int kc = (kb <= 8) ? kb : 8;
      v2f av = *(const v2f*)(x1 + (long)rowc * NCLS + kc);
      float m = ((kb <= 8) ? 1.f : 0.f) * rmask;
      av.x *= m; av.y *= m;
      bfrag[s] = av;
      sq += av.x * av.x + av.y * av.y;
    }
    float sq_tot = sq + __shfl_xor(sq, 16, 32);  // both halves covered k-subsets
    int t = tgt[rowc];                            // same for lane pair

    // ---- 3 chained f32 WMMAs (K = 12, zero padded beyond 10)
    v8f d = {};
#pragma unroll
    for (int s = 0; s < 3; ++s) {
      d = __builtin_amdgcn_wmma_f32_16x16x4_f32(false, afrag[s], false, bfrag[s],
                                                (short)0, d, false, false);
    }

    // ---- lane-local top-2 over this lane's classes; also pick score[target]
    float v1 = NEGINF, v2v = NEGINF, st = 0.f;
    int i1 = 0, i2 = 0;
#pragma unroll
    for (int v = 0; v < 8; ++v) {
      int   c2  = v + 8 * half;
      float val = (c2 < NCLS) ? d[v] : NEGINF;
      st = (c2 == t) ? d[v] : st;
      bool g1 = val > v1;
      bool g2 = val > v2v;
      float nv2 = g1 ? v1 : (g2 ? val : v2v);
      int   ni2 = g1 ? i1 : (g2 ? c2  : i2);
      v1  = g1 ? val : v1;
      i1  = g1 ? c2  : i1;
      v2v = nv2; i2 = ni2;
    }
    // merge with partner half (tie -> lower class index, matching top_k)
    {
      float ov1 = __shfl_xor(v1,  16, 32);
      int   oi1 = __shfl_xor(i1,  16, 32);
      float ov2 = __shfl_xor(v2v, 16, 32);
      int   oi2 = __shfl_xor(i2,  16, 32);
      bool  of  = (ov1 > v1) || ((ov1 == v1) && (oi1 < i1));
      float w1 = of ? ov1 : v1;   int j1 = of ? oi1 : i1;
      float lv = of ? v1  : ov1;  int jl = of ? i1  : oi1;
      float cv = of ? ov2 : v2v;  int jc = of ? oi2 : i2;
      bool  lf = (lv > cv) || ((lv == cv) && (jl < jc));
      v1 = w1; i1 = j1;
      v2v = lf ? lv : cv;
      i2  = lf ? jl : jc;
      st += __shfl_xor(st, 16, 32);  // exactly one lane of the pair matched t
    }

    // ---- per-sample loss (redundant in both lanes of the pair)
    int   mc   = (i1 != t) ? i1 : i2;          // most-confusing class
    float smc  = (i1 != t) ? v1 : v2v;         // its raw score dot(x1, emb[mc])
    float ie_t = __shfl(inv_embn, t,  32);
    float ie_m = __shfl(inv_embn, mc, 32);
    float sqg  = (srow < batch) ? sq_tot : 1.0f;
    float invx = 1.0f / sqrtf(sqg);            // 1/||x1||
    float adv_anchor = 1.0f - st  * invx * ie_t;
    float adv_mc     = 1.0f - smc * invx * ie_m;
    float loss = fmaxf(0.5f - adv_mc, 0.f) + fmaxf(adv_anchor - 0.5f, 0.f);
    acc += loss * rmask;
  }

  // pair lanes hold identical sums: count lower half only, then full-wave sum
  float a = (half == 0) ? acc : 0.f;
#pragma unroll
  for (int off = 1; off < 32; off <<= 1) a += __shfl_xor(a, off, 32);

  __shared__ float wsum[8];
  if (lane == 0) wsum[wid] = a;
  __syncthreads();
  if (threadIdx.x == 0) {
    float s = 0.f;
#pragma unroll
    for (int i = 0; i < 8; ++i) s += wsum[i];
    partials[blockIdx.x] = s;
  }
}

// Deterministic single-block final reduction + mean.
__global__ __launch_bounds__(256) void mcsl_reduce(
    const float* __restrict__ partials, int n, float* __restrict__ out, float invB)
{
  __shared__ float s[256];
  float a = 0.f;
  for (int i = threadIdx.x; i < n; i += 256) a += partials[i];
  s[threadIdx.x] = a;
  __syncthreads();
  for (int st = 128; st > 0; st >>= 1) {
    if ((int)threadIdx.x < st) s[threadIdx.x] += s[threadIdx.x + st];
    __syncthreads();
  }
  if (threadIdx.x == 0) out[0] = s[0] * invB;
}

extern "C" void kernel_launch(void* const* d_in, const int* in_sizes, int n_in,
                              void* d_out, int out_size, void* d_ws, size_t ws_size,
                              hipStream_t stream) {
  const float* x1  = (const float*)d_in[0];
  const int*   tgt = (const int*)d_in[1];
  const float* emb = (const float*)d_in[2];
  const int batch  = in_sizes[1];
  const int ntiles = (batch + 15) / 16;

  int nblk = 1024;  // 8192 waves; ~30 tiles/wave at B=4M; partials = 4 KB of d_ws
  size_t cap = ws_size / sizeof(float);
  if (cap < (size_t)nblk) nblk = (int)(cap ? cap : 1);

  float* partials = (float*)d_ws;
  mcsl_main<<<nblk, 256, 0, stream>>>(x1, tgt, emb, partials, batch, ntiles);
  mcsl_reduce<<<1, 256, 0, stream>>>(partials, nblk, (float*)d_out, 1.0f / (float)batch);
}